// DetectionLoss_81046032875796
// MI455X (gfx1250) — compile-verified
//
#include <hip/hip_runtime.h>
#include <math.h>

// Problem constants (match reference)
#define B_ 64
#define Q_ 4096
#define C_ 80
#define G_ 64
#define M_ 64
#define D_ 7
#define NTOT (64L * 4096L * 80L)   // 20,971,520 score elements
#define NMATCH (B_ * M_)            // 4096 matches
#define EPS_ 1e-8f

#define TPB 256
#define NB_FOCAL 2048               // streaming-kernel blocks (partials in ws)
#define NB_MATCH (NMATCH / TPB)     // 16

typedef __attribute__((ext_vector_type(2))) float v2f;
typedef __attribute__((ext_vector_type(8))) float v8f;

// ---------------------------------------------------------------------------
// Wave32 reduction via V_WMMA_F32_16X16X4_F32 (full-f32, deterministic).
// A layout (ISA 7.12.2, 32-bit 16x4): lane L<16 : VGPR0=A[L][0], VGPR1=A[L][1]
//                                     lane L>=16: VGPR0=A[L-16][2], VGPR1=A[L-16][3]
// Put s in VGPR0, 0 in VGPR1 -> every lane's partial occupies a unique A slot.
// B = all-ones (layout independent).  D[m][n] = s_m + s_{m+16} for all n.
// Per-lane sum of the 8 D VGPRs gives the half-wave total (lanes 0-15 = rows
// 0-7, lanes 16-31 = rows 8-15); shfl_xor(16) combines the halves.
// Requires EXEC all-ones: call only from fully-converged code.
// ---------------------------------------------------------------------------
__device__ __forceinline__ float wave_reduce_wmma(float s) {
  v2f a; a[0] = s;    a[1] = 0.0f;
  v2f b; b[0] = 1.0f; b[1] = 1.0f;
  v8f c = {};
  v8f d = __builtin_amdgcn_wmma_f32_16x16x4_f32(
      /*neg_a=*/false, a, /*neg_b=*/false, b,
      /*c_mod=*/(short)0, c, /*reuse_a=*/false, /*reuse_b=*/false);
  float p = 0.0f;
#pragma unroll
  for (int k = 0; k < 8; ++k) p += d[k];
  p += __shfl_xor(p, 16, 32);
  return p;  // full wave sum, broadcast to all lanes
}

// Deterministic block reduction; result valid on threadIdx.x == 0.
__device__ __forceinline__ float block_reduce_sum(float s, float* lds) {
  const float wt = wave_reduce_wmma(s);
  const int lane = threadIdx.x & 31;
  const int wave = threadIdx.x >> 5;
  const int nw = blockDim.x >> 5;
  if (lane == 0) lds[wave] = wt;
  __syncthreads();
  float tot = 0.0f;
  if (threadIdx.x == 0)
    for (int i = 0; i < nw; ++i) tot += lds[i];
  __syncthreads();  // lds reusable by caller
  return tot;
}

// ---------------------------------------------------------------------------
// Focal loss pieces.  For t=0: ce = max(x,0)+log1p(exp(-|x|)), p_t = 1-p,
// alpha_t = 0.75 -> l0 = 0.75 * p^2 * ce.
// ---------------------------------------------------------------------------
__device__ __forceinline__ float focal0(float x) {
  const float ax = fabsf(x);
  const float e  = __expf(-ax);
  const float lp = __logf(1.0f + e);           // log1p(exp(-|x|))
  const float ce = fmaxf(x, 0.0f) + lp;
  const float inv = __fdividef(1.0f, 1.0f + e);
  const float p  = (x >= 0.0f) ? inv : e * inv; // sigmoid(x), stable
  return 0.75f * p * p * ce;
}

// Correction at a matched element: focal(x, t=1) - focal(x, t=0).
__device__ __forceinline__ float focal_delta(float x) {
  const float ax = fabsf(x);
  const float e  = __expf(-ax);
  const float lp = __logf(1.0f + e);
  const float mx = fmaxf(x, 0.0f);
  const float ce0 = mx + lp;
  const float ce1 = mx - x + lp;
  const float inv = __fdividef(1.0f, 1.0f + e);
  const float p  = (x >= 0.0f) ? inv : e * inv;
  const float om = 1.0f - p;
  return 0.25f * om * om * ce1 - 0.75f * p * p * ce0;
}

// ---------------------------------------------------------------------------
// Kernel 1: stream pred_scores (84 MB) once, sum focal(x, t=0).
// float4 loads, grid-stride; ~3.6 us HBM floor at 23.3 TB/s.
// ---------------------------------------------------------------------------
__global__ void __launch_bounds__(TPB) focal_sum_kernel(
    const float* __restrict__ scores, float* __restrict__ partials) {
  const float4* s4 = (const float4*)scores;
  const int n4 = (int)(NTOT / 4);                 // 5,242,880
  const int stride = gridDim.x * blockDim.x;
  float acc = 0.0f;
  for (int i = blockIdx.x * blockDim.x + threadIdx.x; i < n4; i += stride) {
    __builtin_prefetch(&s4[i + 2 * stride], 0, 3);  // global_prefetch_b8
    const float4 v = s4[i];
    acc += focal0(v.x);
    acc += focal0(v.y);
    acc += focal0(v.z);
    acc += focal0(v.w);
  }
  __shared__ float lds[TPB / 32];
  const float t = block_reduce_sum(acc, lds);
  if (threadIdx.x == 0) partials[blockIdx.x] = t;
}

// ---------------------------------------------------------------------------
// Kernel 2: 4096 matches -> (focal correction, L1 sum, giou-loss sum).
// ---------------------------------------------------------------------------
__global__ void __launch_bounds__(TPB) match_kernel(
    const float* __restrict__ pred_boxes, const float* __restrict__ pred_scores,
    const float* __restrict__ tgt_boxes, const int* __restrict__ tgt_labels,
    const int* __restrict__ pred_indices, const int* __restrict__ gt_indices,
    float* __restrict__ partials) {
  const int j = blockIdx.x * blockDim.x + threadIdx.x;  // exactly 4096 threads
  const int b = j >> 6;                                  // j / M_
  const int pi = pred_indices[j];
  const int gi = gt_indices[j];
  const int cls = tgt_labels[b * G_ + gi];
  const float x = pred_scores[(long)(b * Q_ + pi) * C_ + cls];
  const float dcls = focal_delta(x);

  const float* pb = pred_boxes + (long)(b * Q_ + pi) * D_;
  const float* gb = tgt_boxes + (long)(b * G_ + gi) * D_;
  float pbv[7], gbv[7];
#pragma unroll
  for (int d = 0; d < 7; ++d) { pbv[d] = pb[d]; gbv[d] = gb[d]; }

  float l1 = 0.0f;
#pragma unroll
  for (int d = 0; d < 7; ++d) l1 += fabsf(pbv[d] - gbv[d]);

  // GIoU on dims 0..5 (cx,cy,cz,w,h,l), exactly mirroring the reference math.
  float ivol = 1.0f, evol = 1.0f;
#pragma unroll
  for (int a = 0; a < 3; ++a) {
    const float ph = 0.5f * pbv[a + 3], gh = 0.5f * gbv[a + 3];
    const float pmn = pbv[a] - ph, pmx = pbv[a] + ph;
    const float gmn = gbv[a] - gh, gmx = gbv[a] + gh;
    const float it = fmaxf(fminf(pmx, gmx) - fmaxf(pmn, gmn), 0.0f);
    const float en = fmaxf(fmaxf(pmx, gmx) - fminf(pmn, gmn), 0.0f);
    ivol *= it;
    evol *= en;
  }
  const float pvol = pbv[3] * pbv[4] * pbv[5];
  const float gvol = gbv[3] * gbv[4] * gbv[5];
  const float uni  = pvol + gvol - ivol + EPS_;
  const float iou  = ivol / uni;
  const float encv = evol + EPS_;
  const float giou = iou - (encv - uni) / encv;
  const float gl = 1.0f - giou;

  __shared__ float lds[TPB / 32];
  const float s0 = block_reduce_sum(dcls, lds);
  const float s1 = block_reduce_sum(l1, lds);
  const float s2 = block_reduce_sum(gl, lds);
  if (threadIdx.x == 0) {
    partials[blockIdx.x * 3 + 0] = s0;
    partials[blockIdx.x * 3 + 1] = s1;
    partials[blockIdx.x * 3 + 2] = s2;
  }
}

// ---------------------------------------------------------------------------
// Kernel 3: deterministic finalize (fixed summation order).
// ---------------------------------------------------------------------------
__global__ void __launch_bounds__(TPB) finalize_kernel(
    const float* __restrict__ focal_partials,
    const float* __restrict__ match_partials, float* __restrict__ out) {
  float s = 0.0f;
  for (int i = threadIdx.x; i < NB_FOCAL; i += TPB) s += focal_partials[i];
  __shared__ float lds[TPB / 32];
  const float focal_sum = block_reduce_sum(s, lds);
  if (threadIdx.x == 0) {
    float dcls = 0.0f, sl1 = 0.0f, sgl = 0.0f;
    for (int k = 0; k < NB_MATCH; ++k) {
      dcls += match_partials[3 * k + 0];
      sl1  += match_partials[3 * k + 1];
      sgl  += match_partials[3 * k + 2];
    }
    const float loss_cls  = (focal_sum + dcls) / (float)NTOT;
    const float loss_bbox = sl1 / (float)(NMATCH * D_);
    const float loss_giou = sgl / (float)NMATCH;
    out[0] = 2.0f * loss_cls + 0.25f * loss_bbox + 0.25f * loss_giou;
    out[1] = loss_cls;
    out[2] = loss_bbox;
    out[3] = loss_giou;
  }
}

extern "C" void kernel_launch(void* const* d_in, const int* in_sizes, int n_in,
                              void* d_out, int out_size, void* d_ws,
                              size_t ws_size, hipStream_t stream) {
  (void)in_sizes; (void)n_in; (void)out_size; (void)ws_size;
  const float* pred_boxes   = (const float*)d_in[0];
  const float* pred_scores  = (const float*)d_in[1];
  const float* tgt_boxes    = (const float*)d_in[2];
  const int*   tgt_labels   = (const int*)d_in[3];
  const int*   pred_indices = (const int*)d_in[4];
  const int*   gt_indices   = (const int*)d_in[5];
  float* out = (float*)d_out;
  float* ws  = (float*)d_ws;
  float* focal_partials = ws;             // NB_FOCAL floats
  float* match_partials = ws + NB_FOCAL;  // NB_MATCH * 3 floats

  focal_sum_kernel<<<NB_FOCAL, TPB, 0, stream>>>(pred_scores, focal_partials);
  match_kernel<<<NB_MATCH, TPB, 0, stream>>>(pred_boxes, pred_scores, tgt_boxes,
                                             tgt_labels, pred_indices,
                                             gt_indices, match_partials);
  finalize_kernel<<<1, TPB, 0, stream>>>(focal_partials, match_partials, out);
}